// CLIPMutationLoss_50646254354663
// MI455X (gfx1250) — compile-verified
//
#include <hip/hip_runtime.h>

// Problem constants (from the reference).
#define B_     8
#define NNODES 2048
#define D_     256
#define T_     1024
#define M1     20   // NUM_MUT + 1 classes

typedef __attribute__((ext_vector_type(2))) float v2f;
typedef __attribute__((ext_vector_type(8))) float v8f;

// Workspace layout (floats): [0]=loss_sum, [1]=correct_sum, [2]=mask_sum, [3..10]=row_counts[8]

// ---------------------------------------------------------------------------
// Kernel A: per-batch-row mask counts + zero the global accumulators.
// ---------------------------------------------------------------------------
__global__ void clip_rowcount_kernel(const unsigned char* __restrict__ mask,
                                     float* __restrict__ ws) {
    __shared__ int red[256];
    const int b = blockIdx.x;
    const int tid = threadIdx.x;
    int c = 0;
    for (int t = tid; t < T_; t += 256) c += mask[b * T_ + t] ? 1 : 0;
    red[tid] = c;
    __syncthreads();
    for (int s = 128; s > 0; s >>= 1) {
        if (tid < s) red[tid] += red[tid + s];
        __syncthreads();
    }
    if (tid == 0) {
        ws[3 + b] = (float)red[0];
        if (b == 0) { ws[0] = 0.0f; ws[1] = 0.0f; ws[2] = 0.0f; }
    }
}

// ---------------------------------------------------------------------------
// Kernel B: one wave32 per (batch, 16-token block).
//   - gather 16 gnn rows into LDS (16 KB)
//   - for each of 20 classes: C(16x16) = Text_m(16x256) x G^T(256x16) via
//     64 x v_wmma_f32_16x16x4_f32 ; diag(C) = per-token scores
//   - lanes 0..15: 20-class logsumexp / argmax, masked atomic accumulation
// ---------------------------------------------------------------------------
__global__ void __launch_bounds__(32)
clip_main_kernel(const float* __restrict__ gnn,
                 const float* __restrict__ text,
                 const float* __restrict__ logit_scale,
                 const long long* __restrict__ coords,
                 const unsigned char* __restrict__ mask,
                 float* __restrict__ ws) {
    __shared__ __align__(16) float sG[16 * D_];   // gathered gnn rows (B operand)
    __shared__ float sS[M1 * 16];                 // raw scores [class][token]

    const int tb   = blockIdx.x;       // token block index (16 tokens)
    const int b    = blockIdx.y;       // batch index
    const int lane = threadIdx.x;      // 0..31 (wave32)
    const int j    = lane & 15;        // token-in-block; also A-row / B-col index
    const int sub  = lane >> 4;        // half-wave: selects K sub-pair of the 16x4 tile
    const int t0   = tb * 16;

    // --- Gather: 16 rows x 256 f32, two lanes per row (float4 vectorized). ---
    const long long node = coords[(size_t)b * T_ + (t0 + j)];
    const float4* src = (const float4*)(gnn + ((size_t)b * NNODES + (size_t)node) * D_) + sub * 32;
    float4* dst = (float4*)(&sG[j * D_]) + sub * 32;
#pragma unroll 4
    for (int i = 0; i < 32; ++i) dst[i] = src[i];
    __syncthreads();

    // Branchless diag-extract setup:
    // diag(C) lives at (vgpr r, lane r) for tokens 0-7 and (vgpr r, lane r+24)
    // for tokens 8-15 => lanes 0-7 and 24-31 each own one diagonal element,
    // at accumulator vector index (lane & 7).
    const int  didx    = lane & 7;
    const bool dactive = (lane < 8) || (lane >= 24);
    const int  dtok    = (lane < 8) ? lane : (lane - 16);

    // --- Per class: K=256 accumulated as 64 WMMA f32 16x16x4 steps. ---
    for (int m = 0; m < M1; ++m) {
        // A-layout (16x4 f32): lanes 0-15 hold rows M=0..15 at K={0,1}; lanes
        // 16-31 the same rows at K={2,3}.  -> lane loads float2 at d = kk*4 + sub*2.
        const float* arow = text + (((size_t)b * M1 + m) * T_ + (t0 + j)) * (size_t)D_ + sub * 2;
        const float* brow = &sG[j * D_ + sub * 2];   // B (4x16): same lane mapping, cols = tokens
        v8f acc = {};
#pragma unroll 8
        for (int kk = 0; kk < D_ / 4; ++kk) {
            v2f a = *(const v2f*)(arow + kk * 4);    // streamed text, read exactly once
            v2f g = *(const v2f*)(brow + kk * 4);    // ds_load_b64 from LDS, reused 20x
            acc = __builtin_amdgcn_wmma_f32_16x16x4_f32(
                /*neg_a=*/false, a, /*neg_b=*/false, g,
                /*c_mod=*/(short)0, acc, /*reuse_a=*/false, /*reuse_b=*/false);
        }
        // Branchless select of this lane's diagonal element (7 v_cndmask),
        // then a single predicated ds_store.
        float dval = acc[0];
#pragma unroll
        for (int r = 1; r < 8; ++r) dval = (didx == r) ? acc[r] : dval;
        if (dactive) sS[m * 16 + dtok] = dval;
    }
    __syncthreads();

    // --- Per-token CE loss + accuracy (one token per lane, 16 lanes). ---
    if (lane < 16) {
        const float scale = logit_scale[0];
        const float s0 = sS[lane] * scale;
        float mx = s0;
#pragma unroll
        for (int m = 1; m < M1; ++m) mx = fmaxf(mx, sS[m * 16 + lane] * scale);
        float se = 0.0f;
#pragma unroll
        for (int m = 0; m < M1; ++m) se += expf(sS[m * 16 + lane] * scale - mx);
        const float loss_tok = logf(se) + (mx - s0);      // -log_softmax[class 0]
        const float correct  = (s0 >= mx) ? 1.0f : 0.0f;  // argmax==0 (ties -> first)
        if (mask[(size_t)b * T_ + (t0 + lane)]) {
            const float rc = ws[3 + b];                   // row count (>=1 guaranteed)
            atomicAdd(&ws[0], loss_tok / rc);
            atomicAdd(&ws[1], correct);
            atomicAdd(&ws[2], 1.0f);
        }
    }
}

// ---------------------------------------------------------------------------
// Kernel C: finalize the two scalars.
// ---------------------------------------------------------------------------
__global__ void clip_final_kernel(const float* __restrict__ ws, float* __restrict__ out) {
    if (threadIdx.x == 0 && blockIdx.x == 0) {
        out[0] = ws[0] / (float)B_;   // loss
        out[1] = ws[1] / ws[2];       // accuracy
    }
}

// ---------------------------------------------------------------------------
extern "C" void kernel_launch(void* const* d_in, const int* in_sizes, int n_in,
                              void* d_out, int out_size, void* d_ws, size_t ws_size,
                              hipStream_t stream) {
    const float*         gnn    = (const float*)d_in[0];          // [8,2048,256] f32
    const float*         text   = (const float*)d_in[1];          // [160,1024,256] f32
    const float*         lscale = (const float*)d_in[2];          // [1] f32
    const long long*     coords = (const long long*)d_in[3];      // [8,1024] i64
    const unsigned char* mask   = (const unsigned char*)d_in[4];  // [8,1024] bool
    float* out = (float*)d_out;
    float* ws  = (float*)d_ws;
    (void)in_sizes; (void)n_in; (void)out_size; (void)ws_size;

    clip_rowcount_kernel<<<B_, 256, 0, stream>>>(mask, ws);
    dim3 grid(T_ / 16, B_);                       // 64 token-blocks x 8 batches
    clip_main_kernel<<<grid, 32, 0, stream>>>(gnn, text, lscale, coords, mask, ws);
    clip_final_kernel<<<1, 32, 0, stream>>>(ws, out);
}